// LaplacianSmoothingLoss_40656160424345
// MI455X (gfx1250) — compile-verified
//
#include <hip/hip_runtime.h>

// Laplacian smoothing loss for MI455X (gfx1250, wave32).
// Pipeline:
//   K1: segment-start detection over sorted `row` (no atomics).
//   K2: per-vertex segment mean -> normalize -> |x*100 - mean| -> |.-xyz_dif|,
//       block-reduced to one partial per 256-vertex block (deterministic).
//   K3: final reduction; bulk of it done with V_WMMA_F32_16X16X4_F32
//       (D += ones(16x4) x B(4x16); sum(D) == 16 * sum(B), layout-agnostic).

typedef __attribute__((ext_vector_type(2))) float v2f;
typedef __attribute__((ext_vector_type(8))) float v8f;

__global__ void seg_starts_kernel(const int* __restrict__ row,
                                  int* __restrict__ start,
                                  int ne, int nv) {
    int i = blockIdx.x * blockDim.x + threadIdx.x;
    if (i >= ne) return;
    int r = row[i];
    if (i == 0) {
        start[r]  = 0;   // row sorted, every vertex present -> r == 0
        start[nv] = ne;  // sentinel
    } else if (row[i - 1] != r) {
        start[r] = i;
    }
}

__global__ void vertex_loss_kernel(const float* __restrict__ x,
                                   const int* __restrict__ col,
                                   const int* __restrict__ start,
                                   const float* __restrict__ xyz_dif,
                                   float* __restrict__ partials,
                                   int nv) {
    __shared__ float red[256];
    const int v = blockIdx.x * 256 + threadIdx.x;

    float local = 0.0f;
    if (v < nv) {
        const int s = start[v];
        const int e = start[v + 1];

        // Segment sum of gathered neighbor positions (L2-resident gather:
        // x is 24MB, col is 48MB -> both fit in the 192MB global L2).
        float sx = 0.0f, sy = 0.0f, sz = 0.0f;
        for (int j = s; j < e; ++j) {
            const size_t c = (size_t)col[j] * 3u;
            sx += x[c + 0];
            sy += x[c + 1];
            sz += x[c + 2];
        }

        int cnt = e - s;
        if (cnt < 1) cnt = 1;                 // torch_scatter clamp semantics
        const float inv_cnt = 1.0f / (float)cnt;
        // Note: reference scales x by 100 before the mean, but the subsequent
        // L2-normalization cancels any positive scale, so mean of raw x is
        // equivalent for the normalized direction.
        float mx = sx * inv_cnt;
        float my = sy * inv_cnt;
        float mz = sz * inv_cnt;
        const float nrm = sqrtf(mx * mx + my * my + mz * mz);
        const float sc  = 100.0f / nrm;       // matches reference (inf/nan if nrm==0)
        mx *= sc; my *= sc; mz *= sc;

        const size_t vb = (size_t)v * 3u;
        const float xd0 = fabsf(fabsf(x[vb + 0] * 100.0f - mx) - xyz_dif[vb + 0]);
        const float xd1 = fabsf(fabsf(x[vb + 1] * 100.0f - my) - xyz_dif[vb + 1]);
        const float xd2 = fabsf(fabsf(x[vb + 2] * 100.0f - mz) - xyz_dif[vb + 2]);
        local = xd0 + xd1 + xd2;
    }

    // Deterministic block tree reduction.
    red[threadIdx.x] = local;
    __syncthreads();
    for (int off = 128; off > 0; off >>= 1) {
        if ((int)threadIdx.x < off) red[threadIdx.x] += red[threadIdx.x + off];
        __syncthreads();
    }
    if (threadIdx.x == 0) partials[blockIdx.x] = red[0];
}

__global__ void final_reduce_kernel(const float* __restrict__ partials,
                                    int nparts,
                                    float* __restrict__ out,
                                    float inv_total) {
    __shared__ float buf[256];

    // Strided deterministic accumulation of block partials.
    float acc = 0.0f;
    for (int i = threadIdx.x; i < nparts; i += 256) acc += partials[i];
    buf[threadIdx.x] = acc;
    __syncthreads();

    // Wave 0 (32 lanes, EXEC all-ones as WMMA requires) reduces the 256
    // values with 4 accumulating V_WMMA_F32_16X16X4_F32 ops:
    //   D += ones(16x4) x B(4x16)  =>  sum over entire D == 16 * sum(B tiles).
    // The grand-total identity holds for ANY bijective packing of the 64
    // values into B, so it is robust to VGPR-layout details.
    if (threadIdx.x < 32) {
        v8f d = {};
        v2f a_ones;
        a_ones[0] = 1.0f;
        a_ones[1] = 1.0f;
        for (int t = 0; t < 4; ++t) {
            v2f b;
            b[0] = buf[t * 64 + (int)threadIdx.x];
            b[1] = buf[t * 64 + 32 + (int)threadIdx.x];
            d = __builtin_amdgcn_wmma_f32_16x16x4_f32(
                    /*neg_a=*/false, a_ones,
                    /*neg_b=*/false, b,
                    /*c_mod=*/(short)0, d,
                    /*reuse_a=*/false, /*reuse_b=*/false);
        }
        // Per-lane sum of its 8 D components; all lanes together cover the
        // full 16x16 D matrix exactly once.
        buf[threadIdx.x] =
            d[0] + d[1] + d[2] + d[3] + d[4] + d[5] + d[6] + d[7];
    }
    __syncthreads();

    if (threadIdx.x == 0) {
        float s = 0.0f;
        for (int i = 0; i < 32; ++i) s += buf[i];   // deterministic tail
        out[0] = s * (1.0f / 16.0f) * inv_total;    // /16 rows, /(NV*3) mean
    }
}

extern "C" void kernel_launch(void* const* d_in, const int* in_sizes, int n_in,
                              void* d_out, int out_size, void* d_ws, size_t ws_size,
                              hipStream_t stream) {
    const float* x       = (const float*)d_in[0];
    const int*   row     = (const int*)d_in[1];
    const int*   col     = (const int*)d_in[2];
    const float* xyz_dif = (const float*)d_in[3];

    const int nv = in_sizes[0] / 3;
    const int ne = in_sizes[1];

    // Workspace layout: [ start: (nv+1) ints | partials: nblk2 floats ]
    int* start = (int*)d_ws;
    size_t start_bytes = (((size_t)(nv + 1) * sizeof(int)) + 255u) & ~(size_t)255u;
    float* partials = (float*)((char*)d_ws + start_bytes);

    const int nblk1 = (ne + 255) / 256;
    const int nblk2 = (nv + 255) / 256;

    seg_starts_kernel<<<nblk1, 256, 0, stream>>>(row, start, ne, nv);
    vertex_loss_kernel<<<nblk2, 256, 0, stream>>>(x, col, start, xyz_dif, partials, nv);
    final_reduce_kernel<<<1, 256, 0, stream>>>(partials, nblk2, (float*)d_out,
                                               1.0f / ((float)nv * 3.0f));
}